// VAE_41180146434656
// MI455X (gfx1250) — compile-verified
//
#include <hip/hip_runtime.h>
#include <math.h>

typedef float v2f __attribute__((ext_vector_type(2)));
typedef float v8f __attribute__((ext_vector_type(8)));

#define LOG2PI 1.8378770664093453f

#if defined(__has_builtin)
#if __has_builtin(__builtin_amdgcn_tanhf)
#define TANHF(x) __builtin_amdgcn_tanhf(x)   // native V_TANH_F32 on gfx1250
#endif
#endif
#ifndef TANHF
#define TANHF(x) tanhf(x)
#endif

// ---- LDS layout (float indices), weights pre-padded to WMMA tile geometry ----
#define OFF_EW1 0       // enc_w1 padded [4][112]   448  (rows>=2, cols>=100 are 0)
#define OFF_EB1 448     // enc_b1 padded [112]      112
#define OFF_EW2 560     // enc_w2 padded [104][16] 1664  (cols>=4, rows>=100 are 0)
#define OFF_EB2 2224    // enc_b2 padded [16]        16
#define OFF_DW1 2240    // dec_w1 padded [4][112]   448
#define OFF_DB1 2688    // dec_b1 padded [112]      112
#define OFF_DW2 2800    // dec_w2 padded [104][16] 1664  (cols>=2 are 0)
#define OFF_DB2 4464    // dec_b2 padded [16]        16
#define OFF_FW1 4480    // flow_w1 (10,1,20)        200
#define OFF_FB1 4680    // flow_b1 (10,20)          200
#define OFF_FW2 4880    // flow_w2 (10,20,1)        200
#define OFF_FB2 5080    // flow_b2 (10,1)            10
#define WEIGHT_FLOATS 5090
#define OFF_RED 5090    // 4 wave partials
#define OFF_WAVE 5096   // per-wave scratch
#define HSTRIDE 114     // even -> 8B-aligned K pairs, conflict-free across 16 rows
#define WS_H 0          // 16*114 = 1824
#define WS_Q 1824       // 32*4
#define WS_Z 1952       // 32*2
#define WS_PX 2016      // 32*2
#define WS_SIZE 2080
#define SMEM_FLOATS (OFF_WAVE + 4 * WS_SIZE)   // 13416 floats ~= 52.4 KB

static __device__ __forceinline__ v8f wmma4(v2f a, v2f b, v8f c) {
  // V_WMMA_F32_16X16X4_F32 : D = A(16x4) * B(4x16) + C(16x16)
  return __builtin_amdgcn_wmma_f32_16x16x4_f32(false, a, false, b, (short)0, c,
                                               false, false);
}

// Two-layer MLP (2 -> 100 -> n2) with ReLU on a 16-sample tile.
// All weight arrays are zero-padded in LDS, so every lane loads unconditionally.
static __device__ __forceinline__ void mlp_block(
    const float* __restrict__ w1p,  // [4][112]
    const float* __restrict__ b1p,  // [112]
    const float* __restrict__ w2p,  // [104][16]
    const float* __restrict__ b2p,  // [16]
    int n2, float in0, float in1, float* __restrict__ H,
    float* __restrict__ OUT, int outBase, int lane) {
  const int half = lane >> 4;   // K-half for A/B fragments
  const int col  = lane & 15;   // N column (B/C/D) or M row (layer-2 A)

  // ---- layer 1: 16x112 tile row, K=2 padded to 4 ----
  v2f A;                        // lanes 0..15: K=0,1 ; lanes 16..31: K=2,3 (zeros)
  A.x = half ? 0.0f : in0;
  A.y = half ? 0.0f : in1;
  const int w1row = (2 * half) * 112;       // rows 2,3 are zero-padded
#pragma unroll
  for (int j = 0; j < 7; ++j) {
    const int n = 16 * j + col;
    v2f B;
    B.x = w1p[w1row + n];
    B.y = w1p[w1row + 112 + n];
    const float bias = b1p[n];
    v8f C;
#pragma unroll
    for (int r = 0; r < 8; ++r) C[r] = bias;
    C = wmma4(A, B, C);
#pragma unroll
    for (int r = 0; r < 8; ++r)
      H[(r + 8 * half) * HSTRIDE + n] = fmaxf(C[r], 0.0f);
  }

  // ---- layer 2: 100 -> n2 (N padded to 16), K=100 as 25 chunks of 4 ----
  const float biasO = b2p[col];             // zero beyond n2
  v8f C2;
#pragma unroll
  for (int r = 0; r < 8; ++r) C2[r] = biasO;
#pragma unroll 5
  for (int c = 0; c < 25; ++c) {
    const int k0 = 4 * c;
    const int kA = k0 + 2 * half;
    v2f A2;                                 // A: m = col, K striped by half
    A2.x = H[col * HSTRIDE + kA];
    A2.y = H[col * HSTRIDE + kA + 1];
    v2f B2;                                 // rows k0+2*half, k0+2*half+1
    B2.x = w2p[kA * 16 + col];
    B2.y = w2p[(kA + 1) * 16 + col];
    C2 = wmma4(A2, B2, C2);
  }
  if (col < n2) {
#pragma unroll
    for (int r = 0; r < 8; ++r)
      OUT[(outBase + r + 8 * half) * n2 + col] = C2[r];
  }
}

__global__ __launch_bounds__(128) void vae_flow_kernel(
    const float* __restrict__ x, const float* __restrict__ eps,
    const float* __restrict__ enc_w1, const float* __restrict__ enc_b1,
    const float* __restrict__ enc_w2, const float* __restrict__ enc_b2,
    const float* __restrict__ dec_w1, const float* __restrict__ dec_b1,
    const float* __restrict__ dec_w2, const float* __restrict__ dec_b2,
    const float* __restrict__ fw1, const float* __restrict__ fb1,
    const float* __restrict__ fw2, const float* __restrict__ fb2,
    float* __restrict__ zout /* d_out+1 */, float* __restrict__ partials) {
  __shared__ float sm[SMEM_FLOATS];
  const int tid = threadIdx.x;

  // ---- zero the padded weight region, then scatter the real weights ----
  for (int i = tid; i < WEIGHT_FLOATS; i += 128) sm[i] = 0.0f;
  __syncthreads();
  for (int i = tid; i < 200; i += 128)
    sm[OFF_EW1 + (i / 100) * 112 + (i % 100)] = enc_w1[i];
  for (int i = tid; i < 100; i += 128) sm[OFF_EB1 + i] = enc_b1[i];
  for (int i = tid; i < 400; i += 128)
    sm[OFF_EW2 + (i >> 2) * 16 + (i & 3)] = enc_w2[i];
  if (tid < 4) sm[OFF_EB2 + tid] = enc_b2[tid];
  for (int i = tid; i < 200; i += 128)
    sm[OFF_DW1 + (i / 100) * 112 + (i % 100)] = dec_w1[i];
  for (int i = tid; i < 100; i += 128) sm[OFF_DB1 + i] = dec_b1[i];
  for (int i = tid; i < 200; i += 128)
    sm[OFF_DW2 + (i >> 1) * 16 + (i & 1)] = dec_w2[i];
  if (tid < 2) sm[OFF_DB2 + tid] = dec_b2[tid];
  for (int i = tid; i < 200; i += 128) {
    sm[OFF_FW1 + i] = fw1[i];
    sm[OFF_FB1 + i] = fb1[i];
    sm[OFF_FW2 + i] = fw2[i];
  }
  if (tid < 10) sm[OFF_FB2 + tid] = fb2[tid];
  __syncthreads();

  const int w    = tid >> 5;
  const int lane = tid & 31;
  const int half = lane >> 4;
  float* WSM = sm + OFF_WAVE + w * WS_SIZE;
  float* H  = WSM + WS_H;
  float* Q  = WSM + WS_Q;
  float* Z  = WSM + WS_Z;
  float* PX = WSM + WS_PX;
  const int waveBase = blockIdx.x * 128 + w * 32;

  // ---- encoder: two 16-sample M-tiles per wave ----
  for (int T = 0; T < 2; ++T) {
    float in0 = 0.0f, in1 = 0.0f;
    if (!half) {
      const int s = waveBase + T * 16 + lane;
      in0 = x[2 * s];
      in1 = x[2 * s + 1];
    }
    mlp_block(sm + OFF_EW1, sm + OFF_EB1, sm + OFF_EW2, sm + OFF_EB2, 4, in0,
              in1, H, Q, T * 16, lane);
  }

  // ---- per-lane sample: sampling, logqz, flows ----
  const int s = waveBase + lane;
  const float x0 = x[2 * s], x1 = x[2 * s + 1];
  const float e0 = eps[2 * s], e1 = eps[2 * s + 1];
  const float m0 = Q[lane * 4 + 0], m1 = Q[lane * 4 + 1];
  const float l0 = Q[lane * 4 + 2], l1 = Q[lane * 4 + 3];
  float z0 = fmaf(e0, __expf(l0), m0);
  float z1 = fmaf(e1, __expf(l1), m1);
  const float t0 = (z0 - m0) * __expf(-l0);
  const float t1 = (z1 - m1) * __expf(-l1);
  const float logqz = -0.5f * (t0 * t0 + 2.0f * l0 + LOG2PI)
                      -0.5f * (t1 * t1 + 2.0f * l1 + LOG2PI);

  for (int i = 0; i < 10; ++i) {
    const float u = (i & 1) ? z1 : z0;
    float acc = sm[OFF_FB2 + i];
#pragma unroll
    for (int j = 0; j < 20; ++j) {
      const float t = TANHF(fmaf(u, sm[OFF_FW1 + i * 20 + j],
                                 sm[OFF_FB1 + i * 20 + j]));
      acc = fmaf(t, sm[OFF_FW2 + i * 20 + j], acc);
    }
    if (i & 1) z0 += acc; else z1 += acc;
  }
  const float logpz = -0.5f * (z0 * z0 + LOG2PI) - 0.5f * (z1 * z1 + LOG2PI);

  Z[lane * 2 + 0] = z0;
  Z[lane * 2 + 1] = z1;
  zout[2 * s]     = z0;
  zout[2 * s + 1] = z1;

  // ---- decoder on z ----
  for (int T = 0; T < 2; ++T) {
    float in0 = 0.0f, in1 = 0.0f;
    if (!half) {
      in0 = Z[(T * 16 + lane) * 2];
      in1 = Z[(T * 16 + lane) * 2 + 1];
    }
    mlp_block(sm + OFF_DW1, sm + OFF_DB1, sm + OFF_DW2, sm + OFF_DB2, 2, in0,
              in1, H, PX, T * 16, lane);
  }
  const float p0 = PX[lane * 2 + 0], p1 = PX[lane * 2 + 1];
  const float d0 = x0 - p0, d1 = x1 - p1;
  const float logpxz = -0.5f * (d0 * d0 + LOG2PI) - 0.5f * (d1 * d1 + LOG2PI);

  // ---- deterministic reduction: wave -> block partial ----
  float val = logpxz + logpz - logqz;
#pragma unroll
  for (int off = 16; off > 0; off >>= 1) val += __shfl_xor(val, off, 32);
  if (lane == 0) sm[OFF_RED + w] = val;
  __syncthreads();
  if (tid == 0)
    partials[blockIdx.x] =
        (sm[OFF_RED + 0] + sm[OFF_RED + 1]) + (sm[OFF_RED + 2] + sm[OFF_RED + 3]);
}

__global__ __launch_bounds__(256) void elbo_reduce_kernel(
    const float* __restrict__ partials, int n, float* __restrict__ out,
    float invN) {
  __shared__ float s[256];
  float acc = 0.0f;
  for (int i = threadIdx.x; i < n; i += 256) acc += partials[i];  // fixed order
  s[threadIdx.x] = acc;
  __syncthreads();
  for (int off = 128; off > 0; off >>= 1) {
    if (threadIdx.x < off) s[threadIdx.x] += s[threadIdx.x + off];
    __syncthreads();
  }
  if (threadIdx.x == 0) out[0] = s[0] * invN;
}

extern "C" void kernel_launch(void* const* d_in, const int* in_sizes, int n_in,
                              void* d_out, int out_size, void* d_ws,
                              size_t ws_size, hipStream_t stream) {
  const float* x      = (const float*)d_in[0];
  const float* eps    = (const float*)d_in[1];
  const float* enc_w1 = (const float*)d_in[2];
  const float* enc_b1 = (const float*)d_in[3];
  const float* enc_w2 = (const float*)d_in[4];
  const float* enc_b2 = (const float*)d_in[5];
  const float* dec_w1 = (const float*)d_in[6];
  const float* dec_b1 = (const float*)d_in[7];
  const float* dec_w2 = (const float*)d_in[8];
  const float* dec_b2 = (const float*)d_in[9];
  const float* fw1    = (const float*)d_in[10];
  const float* fb1    = (const float*)d_in[11];
  const float* fw2    = (const float*)d_in[12];
  const float* fb2    = (const float*)d_in[13];

  float* out = (float*)d_out;            // [0]=elbo, [1..2N]=z
  float* partials = (float*)d_ws;

  const int N = in_sizes[0] / 2;         // 1048576
  const int blocks = N / 128;            // 8192 (exact)

  vae_flow_kernel<<<blocks, 128, 0, stream>>>(
      x, eps, enc_w1, enc_b1, enc_w2, enc_b2, dec_w1, dec_b1, dec_w2, dec_b2,
      fw1, fb1, fw2, fb2, out + 1, partials);
  elbo_reduce_kernel<<<1, 256, 0, stream>>>(partials, blocks, out,
                                            1.0f / (float)N);
}